// MoonElecEmb_20031727469144
// MI455X (gfx1250) — compile-verified
//
#include <hip/hip_runtime.h>
#include <hip/hip_bf16.h>
#include <math.h>

// ---------- types ----------
typedef __attribute__((ext_vector_type(2))) float v2f;
typedef __attribute__((ext_vector_type(8))) float v8f;

// D = A(16x4,f32) * B(4x16,f32) + C(16x16,f32)  -- CDNA5 V_WMMA_F32_16X16X4_F32
__device__ __forceinline__ v8f wmma_f32_16x16x4(v2f a, v2f b, v8f c) {
  return __builtin_amdgcn_wmma_f32_16x16x4_f32(
      /*neg_a=*/false, a, /*neg_b=*/false, b,
      /*c_mod=*/(short)0, c, /*reuse_a=*/false, /*reuse_b=*/false);
}

// ---------- fast transcendentals (single trans-unit instructions) ----------
__device__ __forceinline__ float fast_rcp(float x) {
  return __builtin_amdgcn_rcpf(x);            // v_rcp_f32
}
__device__ __forceinline__ float fast_silu(float z) {
  return z * fast_rcp(1.0f + __expf(-z));     // v_exp_f32 + v_rcp_f32
}
__device__ __forceinline__ float fast_tanh(float x) {
#if __has_builtin(__builtin_amdgcn_tanhf)
  return __builtin_amdgcn_tanhf(x);           // v_tanh_f32 (gfx1250 trans op)
#else
  const float e = __expf(2.0f * x);
  return 1.0f - 2.0f * fast_rcp(e + 1.0f);
#endif
}

#define CUTOFF   5.0f
#define N_ENV    8
#define F0       16
#define F1       32
#define DDIM     256
#define NNB      32
#define NSITES   8192   // B*E = 128*64

// =====================================================================
// Kernel A: fully fused pairwise pipeline, one block (8 waves) per site.
// Writes pre-activation res[site][256] to `res_out` (== d_out).
// =====================================================================
__global__ __launch_bounds__(256)
void fused_pair_kernel(const float* __restrict__ r,
                       const float* __restrict__ r_nb,
                       const float* __restrict__ ee_scales,
                       const float* __restrict__ ee_kernel,    // [4,16]
                       const float* __restrict__ ee_bias,      // [16]
                       const float* __restrict__ beta_kernel,  // [24,32]
                       const float* __restrict__ beta_bias,    // [32]
                       const float* __restrict__ gamma_kernel, // [32,256]
                       const float* __restrict__ dense1_kernel,// [4,256]
                       const float* __restrict__ dense1_bias,  // [256]
                       float* __restrict__ res_out)            // [8192,256]
{
  __shared__ float phS[NNB][F0 + N_ENV];   // concat(h, env): [32][24]
  __shared__ float inpS[NNB][4];           // rescaled features
  __shared__ float fcutS[NNB];
  __shared__ float betaS[NNB][F1 + 1];     // [32][33] padded (bank-safe)

  const int be  = blockIdx.x;
  const int tid = threadIdx.x;

  // ---- Phase 1: per-neighbor scalar features (lanes of wave 0) ----
  if (tid < NNB) {
    const int n = tid;
    const float rx = r[be * 3 + 0], ry = r[be * 3 + 1], rz = r[be * 3 + 2];
    const float* nb = r_nb + ((size_t)be * NNB + n) * 3;
    const float dx = nb[0] - rx, dy = nb[1] - ry, dz = nb[2] - rz;
    const float dist = sqrtf(dx * dx + dy * dy + dz * dz);

    #pragma unroll
    for (int j = 0; j < F0; ++j) {
      float acc = ee_bias[j];
      acc += dist * ee_kernel[0 * F0 + j];
      acc += dx   * ee_kernel[1 * F0 + j];
      acc += dy   * ee_kernel[2 * F0 + j];
      acc += dz   * ee_kernel[3 * F0 + j];
      phS[n][j] = fast_tanh(acc);
    }
    #pragma unroll
    for (int s = 0; s < N_ENV; ++s) {
      const float t = dist * fast_rcp(ee_scales[s]);
      phS[n][F0 + s] = __expf(-t * t);
    }
    const float x = dist * (1.0f / CUTOFF);
    fcutS[n] = (x < 1.0f) ? 0.5f * (__cosf((float)M_PI * x) + 1.0f) : 0.0f;

    const float l1   = __logf(1.0f + dist);   // log1p(dist)
    const float linv = l1 * fast_rcp(dist);
    inpS[n][0] = l1;           // (dist/dist)*log1p(dist)
    inpS[n][1] = dx * linv;
    inpS[n][2] = dy * linv;
    inpS[n][3] = dz * linv;
  }
  __syncthreads();

  // ---- Phase 2: beta = (concat(h,env) @ beta_kernel + bias) * fcut ----
  // 256 threads x 4 entries = 32x32
  {
    const int n  = tid >> 3;
    const int fb = (tid & 7) * 4;
    const float fc = fcutS[n];
    #pragma unroll
    for (int j = 0; j < 4; ++j) {
      const int f = fb + j;
      float acc = beta_bias[f];
      #pragma unroll
      for (int k = 0; k < F0 + N_ENV; ++k)
        acc += phS[n][k] * beta_kernel[k * F1 + f];
      betaS[n][f] = acc * fc;
    }
  }
  __syncthreads();

  // ---- Phase 3: gamma = beta @ gamma_kernel via WMMA, fused with
  //      feat = silu(inp @ dense1 + b) and the neighbor reduction. ----
  const int wave  = tid >> 5;
  const int lane  = tid & 31;
  const int mcol  = lane & 15;   // A row / B-C column within 16-tile
  const int khalf = lane >> 4;   // which K pair this lane holds

  // A fragments (beta rows) for both M tiles, all 8 K-steps
  v2f afrag[2][8];
  #pragma unroll
  for (int mt = 0; mt < 2; ++mt) {
    const float* row = &betaS[mt * 16 + mcol][0];
    #pragma unroll
    for (int kk = 0; kk < 8; ++kk) {
      afrag[mt][kk].x = row[kk * 4 + khalf * 2 + 0];
      afrag[mt][kk].y = row[kk * 4 + khalf * 2 + 1];
    }
  }

  const int dbase = wave * 32;
  #pragma unroll
  for (int nt = 0; nt < 2; ++nt) {
    const int d = dbase + nt * 16 + mcol;

    v2f bfrag[8];
    #pragma unroll
    for (int kk = 0; kk < 8; ++kk) {
      bfrag[kk].x = gamma_kernel[(kk * 4 + khalf * 2 + 0) * DDIM + d];
      bfrag[kk].y = gamma_kernel[(kk * 4 + khalf * 2 + 1) * DDIM + d];
    }
    const float w0 = dense1_kernel[0 * DDIM + d];
    const float w1 = dense1_kernel[1 * DDIM + d];
    const float w2 = dense1_kernel[2 * DDIM + d];
    const float w3 = dense1_kernel[3 * DDIM + d];
    const float bb = dense1_bias[d];

    float partial = 0.0f;
    #pragma unroll
    for (int mt = 0; mt < 2; ++mt) {
      v8f c = {};
      #pragma unroll
      for (int kk = 0; kk < 8; ++kk)
        c = wmma_f32_16x16x4(afrag[mt][kk], bfrag[kk], c);
      // c[rr] = gamma[neighbor = mt*16 + khalf*8 + rr][d]
      #pragma unroll
      for (int rr = 0; rr < 8; ++rr) {
        const int n = mt * 16 + khalf * 8 + rr;
        const float z = inpS[n][0] * w0 + inpS[n][1] * w1 +
                        inpS[n][2] * w2 + inpS[n][3] * w3 + bb;
        partial += fast_silu(z) * c[rr];
      }
    }
    // lanes L and L+16 hold complementary neighbor halves of column d
    const float other = __shfl_xor(partial, 16, 32);
    if (khalf == 0)
      res_out[(size_t)be * DDIM + d] = partial + other;
  }
}

// =====================================================================
// Kernel B: res = silu(res @ out_kernel + out_bias), in place over d_out.
// One block = 32 rows x 256 cols; row tile staged in LDS (stride 260).
// =====================================================================
__global__ __launch_bounds__(256)
void out_gemm_kernel(const float* __restrict__ Wk,   // [256,256]
                     const float* __restrict__ bias, // [256]
                     float* __restrict__ io)         // [8192,256] in/out
{
  __shared__ float rowS[32 * 260];   // padded: (m*260+k)%64 conflict-free
  const int m0  = blockIdx.x * 32;
  const int tid = threadIdx.x;

  for (int i = tid; i < 32 * DDIM; i += 256) {
    const int rr = i >> 8, col = i & 255;
    rowS[rr * 260 + col] = io[(size_t)(m0 + rr) * DDIM + col];
  }
  __syncthreads();

  const int wave  = tid >> 5;
  const int lane  = tid & 31;
  const int mcol  = lane & 15;
  const int khalf = lane >> 4;

  #pragma unroll
  for (int nt = 0; nt < 2; ++nt) {
    const int dcol = wave * 32 + nt * 16 + mcol;
    const float bb = bias[dcol];
    #pragma unroll
    for (int mt = 0; mt < 2; ++mt) {
      v8f c = {};
      const float* arow = &rowS[(mt * 16 + mcol) * 260 + khalf * 2];
      #pragma unroll 8
      for (int kk = 0; kk < 64; ++kk) {
        v2f a, b;
        a.x = arow[kk * 4 + 0];
        a.y = arow[kk * 4 + 1];
        b.x = Wk[(kk * 4 + khalf * 2 + 0) * DDIM + dcol];
        b.y = Wk[(kk * 4 + khalf * 2 + 1) * DDIM + dcol];
        c = wmma_f32_16x16x4(a, b, c);
      }
      #pragma unroll
      for (int rr = 0; rr < 8; ++rr) {
        const int row = m0 + mt * 16 + khalf * 8 + rr;
        io[(size_t)row * DDIM + dcol] = fast_silu(c[rr] + bb);
      }
    }
  }
}

// =====================================================================
extern "C" void kernel_launch(void* const* d_in, const int* in_sizes, int n_in,
                              void* d_out, int out_size, void* d_ws, size_t ws_size,
                              hipStream_t stream) {
  const float* r             = (const float*)d_in[0];
  const float* r_nb          = (const float*)d_in[1];
  const float* ee_scales     = (const float*)d_in[2];
  const float* ee_kernel     = (const float*)d_in[3];
  const float* ee_bias       = (const float*)d_in[4];
  const float* beta_kernel   = (const float*)d_in[5];
  const float* beta_bias     = (const float*)d_in[6];
  const float* gamma_kernel  = (const float*)d_in[7];
  const float* dense1_kernel = (const float*)d_in[8];
  const float* dense1_bias   = (const float*)d_in[9];
  const float* out_kernel    = (const float*)d_in[10];
  const float* out_bias      = (const float*)d_in[11];
  float* out = (float*)d_out;

  fused_pair_kernel<<<NSITES, 256, 0, stream>>>(
      r, r_nb, ee_scales, ee_kernel, ee_bias, beta_kernel, beta_bias,
      gamma_kernel, dense1_kernel, dense1_bias, out);

  out_gemm_kernel<<<NSITES / 32, 256, 0, stream>>>(out_kernel, out_bias, out);
}